// ImportGNN_29326036697815
// MI455X (gfx1250) — compile-verified
//
#include <hip/hip_runtime.h>

typedef float v2f __attribute__((ext_vector_type(2)));
typedef float v8f __attribute__((ext_vector_type(8)));

#define N_NODES  262144
#define N_EDGES  4194304
#define N_GRAPHS 1024
#define IN_DIM   10
#define HIDDEN   32

__device__ __forceinline__ void atom_add_f32(float* p, float v) {
    // lowers to global_atomic_add_f32 (no CAS loop)
    unsafeAtomicAdd(p, v);
}

// ---- degree / normalization -------------------------------------------------

__global__ void k_init_deg(float* __restrict__ deg) {
    int i = blockIdx.x * blockDim.x + threadIdx.x;
    if (i < N_NODES) deg[i] = 1.0f;   // self-loop contributes 1
}

__global__ void k_degree(const long long* __restrict__ dst, float* __restrict__ deg) {
    int e = blockIdx.x * blockDim.x + threadIdx.x;
    if (e < N_EDGES) {
        int d = (int)dst[e];
        atom_add_f32(&deg[d], 1.0f);
    }
}

__global__ void k_dinv(float* __restrict__ deg) {
    int i = blockIdx.x * blockDim.x + threadIdx.x;
    if (i < N_NODES) {
        float d = deg[i];
        deg[i] = (d > 0.0f) ? (1.0f / sqrtf(d)) : 0.0f;
    }
}

// ---- dense transform via V_WMMA_F32_16X16X4_F32 -----------------------------
// One wave computes a 16x32 output tile: D = A[16xK] @ W[Kx32], then scales by
// dinv[row] and writes both g (gather source) and acc (scatter accumulator,
// seeded with the self-loop term g[i]).
//
// K is a template constant: full k-steps emit unguarded (mergeable) loads; the
// tail step (K=10 only) uses address-clamp + value-select, which lowers to an
// unconditional load + v_cndmask instead of EXEC-mask branches.
template <int K>
__global__ __launch_bounds__(128) void k_gemm_scale(
        const float* __restrict__ A, const float* __restrict__ W,
        const float* __restrict__ dinv, float* __restrict__ g,
        float* __restrict__ acc) {
    const int lane    = threadIdx.x & 31;
    const int wave    = threadIdx.x >> 5;
    const int rowBase = blockIdx.x * 64 + wave * 16;
    const int m       = lane & 15;                 // A row / B col within tile
    const int khalf   = (lane >> 4) << 1;          // 0 for lanes 0-15, 2 for 16-31
    const int row     = rowBase + m;

    v8f c0 = {};
    v8f c1 = {};
#pragma unroll
    for (int k0 = 0; k0 < K; k0 += 4) {
        const int kb = k0 + khalf;
        v2f a, b0, b1;
        if (k0 + 4 <= K) {
            // statically full step: unguarded loads (A pair merges to b64)
            a.x  = A[row * K + kb];
            a.y  = A[row * K + kb + 1];
            b0.x = W[kb * HIDDEN + m];
            b0.y = W[(kb + 1) * HIDDEN + m];
            b1.x = W[kb * HIDDEN + 16 + m];
            b1.y = W[(kb + 1) * HIDDEN + 16 + m];
        } else {
            // tail: clamp address (always valid), select value -> branchless
            const int k0c = (kb     < K) ? kb     : 0;
            const int k1c = (kb + 1 < K) ? kb + 1 : 0;
            const float a0  = A[row * K + k0c];
            const float a1  = A[row * K + k1c];
            const float w00 = W[k0c * HIDDEN + m];
            const float w01 = W[k1c * HIDDEN + m];
            const float w10 = W[k0c * HIDDEN + 16 + m];
            const float w11 = W[k1c * HIDDEN + 16 + m];
            a.x  = (kb     < K) ? a0  : 0.0f;
            a.y  = (kb + 1 < K) ? a1  : 0.0f;
            b0.x = (kb     < K) ? w00 : 0.0f;
            b0.y = (kb + 1 < K) ? w01 : 0.0f;
            b1.x = (kb     < K) ? w10 : 0.0f;
            b1.y = (kb + 1 < K) ? w11 : 0.0f;
        }
        c0 = __builtin_amdgcn_wmma_f32_16x16x4_f32(false, a, false, b0,
                                                   (short)0, c0, false, false);
        c1 = __builtin_amdgcn_wmma_f32_16x16x4_f32(false, a, false, b1,
                                                   (short)0, c1, false, false);
    }

#pragma unroll
    for (int r = 0; r < 8; ++r) {
        const int rl   = (lane < 16) ? r : r + 8;  // D layout: M=r / M=r+8
        const int orow = rowBase + rl;
        const float s  = dinv[orow];
        const float v0 = c0[r] * s;
        const float v1 = c1[r] * s;
        g[orow * HIDDEN + m]        = v0;
        g[orow * HIDDEN + 16 + m]   = v1;
        acc[orow * HIDDEN + m]      = v0;
        acc[orow * HIDDEN + 16 + m] = v1;
    }
}

// ---- message passing: acc[dst] += g[src] ------------------------------------
// 8 lanes per edge, float4 each -> one 128B line gathered, one line of atomics.
__global__ void k_edge_scatter(const long long* __restrict__ src,
                               const long long* __restrict__ dst,
                               const float* __restrict__ g,
                               float* __restrict__ acc) {
    long long tid = (long long)blockIdx.x * blockDim.x + threadIdx.x;
    int e   = (int)(tid >> 3);
    int sub = (int)(tid & 7);
    if (e < N_EDGES) {
        int s = (int)src[e];
        int d = (int)dst[e];
        const float4 v = *(const float4*)(g + (size_t)s * HIDDEN + sub * 4);
        float* p = acc + (size_t)d * HIDDEN + sub * 4;
        atom_add_f32(p + 0, v.x);
        atom_add_f32(p + 1, v.y);
        atom_add_f32(p + 2, v.z);
        atom_add_f32(p + 3, v.w);
    }
}

// ---- h = relu(dinv[i]*acc + b) ---------------------------------------------
__global__ void k_finalize(const float* __restrict__ acc,
                           const float* __restrict__ dinv,
                           const float* __restrict__ bias,
                           float* __restrict__ h) {
    int t = blockIdx.x * blockDim.x + threadIdx.x;   // N_NODES*HIDDEN threads
    int i = t >> 5;
    int c = t & 31;
    float v = dinv[i] * acc[t] + bias[c];
    h[t] = (v > 0.0f) ? v : 0.0f;
}

// ---- global mean pool -------------------------------------------------------

__global__ void k_pool_init(float* __restrict__ sums, float* __restrict__ cnt) {
    int i = blockIdx.x * blockDim.x + threadIdx.x;
    if (i < N_GRAPHS * HIDDEN) sums[i] = 0.0f;
    if (i < N_GRAPHS) cnt[i] = 0.0f;
}

__global__ void k_pool(const float* __restrict__ h,
                       const long long* __restrict__ batch,
                       float* __restrict__ sums, float* __restrict__ cnt) {
    long long tid = (long long)blockIdx.x * blockDim.x + threadIdx.x;
    int i   = (int)(tid >> 3);
    int sub = (int)(tid & 7);
    if (i < N_NODES) {
        int gph = (int)batch[i];
        const float4 v = *(const float4*)(h + (size_t)i * HIDDEN + sub * 4);
        float* p = sums + gph * HIDDEN + sub * 4;
        atom_add_f32(p + 0, v.x);
        atom_add_f32(p + 1, v.y);
        atom_add_f32(p + 2, v.z);
        atom_add_f32(p + 3, v.w);
        if (sub == 0) atom_add_f32(&cnt[gph], 1.0f);
    }
}

// ---- out[g] = (sums[g]/max(cnt,1)) . Wlin + blin ---------------------------
__global__ void k_final(const float* __restrict__ sums, const float* __restrict__ cnt,
                        const float* __restrict__ Wlin, const float* __restrict__ blin,
                        float* __restrict__ out) {
    int gph  = blockIdx.x * (blockDim.x >> 5) + (threadIdx.x >> 5);
    int lane = threadIdx.x & 31;
    float cn = cnt[gph];
    cn = (cn > 1.0f) ? cn : 1.0f;
    float v = (sums[gph * HIDDEN + lane] / cn) * Wlin[lane];
#pragma unroll
    for (int off = 16; off > 0; off >>= 1) v += __shfl_down(v, off, 32);
    if (lane == 0) out[gph] = v + blin[0];
}

extern "C" void kernel_launch(void* const* d_in, const int* in_sizes, int n_in,
                              void* d_out, int out_size, void* d_ws, size_t ws_size,
                              hipStream_t stream) {
    const float*     x    = (const float*)d_in[0];
    const long long* ei   = (const long long*)d_in[1];   // [2, E] int64
    const long long* batch= (const long long*)d_in[2];
    const float*     W1   = (const float*)d_in[3];
    const float*     b1   = (const float*)d_in[4];
    const float*     W2   = (const float*)d_in[5];
    const float*     b2   = (const float*)d_in[6];
    const float*     Wlin = (const float*)d_in[7];
    const float*     blin = (const float*)d_in[8];
    float*           out  = (float*)d_out;

    const long long* e_src = ei;
    const long long* e_dst = ei + N_EDGES;

    float* dinv = (float*)d_ws;                                // N
    float* g    = dinv + N_NODES;                              // N*32
    float* acc  = g    + (size_t)N_NODES * HIDDEN;             // N*32
    float* sums = acc  + (size_t)N_NODES * HIDDEN;             // 1024*32
    float* cnt  = sums + N_GRAPHS * HIDDEN;                    // 1024

    // normalization: deg -> dinv (in place)
    k_init_deg<<<N_NODES / 256, 256, 0, stream>>>(dinv);
    k_degree<<<N_EDGES / 256, 256, 0, stream>>>(e_dst, dinv);
    k_dinv<<<N_NODES / 256, 256, 0, stream>>>(dinv);

    // layer 1: g = (x@W1)*dinv, acc = g ; acc[dst] += g[src] ; h1 = relu(dinv*acc+b1)
    k_gemm_scale<IN_DIM><<<N_NODES / 64, 128, 0, stream>>>(x, W1, dinv, g, acc);
    k_edge_scatter<<<(N_EDGES * 8) / 256, 256, 0, stream>>>(e_src, e_dst, g, acc);
    k_finalize<<<(N_NODES * HIDDEN) / 256, 256, 0, stream>>>(acc, dinv, b1, g); // h1 in g

    // layer 2 (in-place GEMM on g is safe: each wave reads only rows it writes)
    k_gemm_scale<HIDDEN><<<N_NODES / 64, 128, 0, stream>>>(g, W2, dinv, g, acc);
    k_edge_scatter<<<(N_EDGES * 8) / 256, 256, 0, stream>>>(e_src, e_dst, g, acc);
    k_finalize<<<(N_NODES * HIDDEN) / 256, 256, 0, stream>>>(acc, dinv, b2, g); // h2 in g

    // mean pool + linear head
    k_pool_init<<<(N_GRAPHS * HIDDEN) / 256, 256, 0, stream>>>(sums, cnt);
    k_pool<<<(N_NODES * 8) / 256, 256, 0, stream>>>(g, batch, sums, cnt);
    k_final<<<N_GRAPHS / 8, 256, 0, stream>>>(sums, cnt, Wlin, blin, out);
}